// SampleAndGroup_62921270886557
// MI455X (gfx1250) — compile-verified
//
#include <hip/hip_runtime.h>
#include <cstdint>

#define B_      4
#define N_      8192
#define C_      256
#define S_      1024
#define K_      64
#define RADIUS2 (0.2f * 0.2f)

typedef __attribute__((ext_vector_type(2))) float v2f;
typedef __attribute__((ext_vector_type(8))) float v8f;

// ---------------------------------------------------------------------------
// Kernel 1: farthest point sampling. One block per batch; xyz staged in LDS
// (SoA, 96KB of the 320KB WGP LDS); min-distance state lives in registers
// (32 points/thread, strided so LDS reads are bank-conflict free and the
// lowest-index tie-break matches jnp.argmax's first-max semantics).
// ---------------------------------------------------------------------------
__global__ __launch_bounds__(256) void fps_kernel(const float* __restrict__ xyz,
                                                  float* __restrict__ sample_xyz)
{
    __shared__ float sx[N_], sy[N_], sz[N_];
    __shared__ float rv[8];
    __shared__ int   ri[8];
    __shared__ int   s_cur;

    const int b    = blockIdx.x;
    const int tid  = threadIdx.x;
    const int lane = tid & 31;
    const int wave = tid >> 5;

    const float* src = xyz + (size_t)b * N_ * 3;
    for (int i = tid; i < N_ * 3; i += 256) {
        float v = src[i];
        int n = i / 3, d = i - n * 3;
        if      (d == 0) sx[n] = v;
        else if (d == 1) sy[n] = v;
        else             sz[n] = v;
    }
    __syncthreads();

    float min_d[32];
    #pragma unroll
    for (int j = 0; j < 32; ++j) min_d[j] = 3.4e38f;

    int cur = 0;  // deterministic start at index 0, like the reference
    for (int it = 0; it < S_; ++it) {
        if (tid == 0) {
            float* o = sample_xyz + ((size_t)b * S_ + it) * 3;
            o[0] = sx[cur]; o[1] = sy[cur]; o[2] = sz[cur];
        }
        const float cx = sx[cur], cy = sy[cur], cz = sz[cur];

        float best = -1.0f; int bidx = 0x7fffffff;
        #pragma unroll
        for (int j = 0; j < 32; ++j) {
            int n = tid + j * 256;                 // strided ownership
            float dx = sx[n] - cx, dy = sy[n] - cy, dz = sz[n] - cz;
            float d  = dx * dx + dy * dy + dz * dz;
            float md = fminf(min_d[j], d);
            min_d[j] = md;
            if (md > best) { best = md; bidx = n; }
        }
        // wave32 butterfly reduction: max value, lowest index on tie
        #pragma unroll
        for (int off = 16; off >= 1; off >>= 1) {
            float ov = __shfl_xor(best, off);
            int   oi = __shfl_xor(bidx, off);
            if (ov > best || (ov == best && oi < bidx)) { best = ov; bidx = oi; }
        }
        if (lane == 0) { rv[wave] = best; ri[wave] = bidx; }
        __syncthreads();
        if (wave == 0) {
            float v = (lane < 8) ? rv[lane] : -1.0f;
            int   i = (lane < 8) ? ri[lane] : 0x7fffffff;
            #pragma unroll
            for (int off = 4; off >= 1; off >>= 1) {
                float ov = __shfl_xor(v, off);
                int   oi = __shfl_xor(i, off);
                if (ov > v || (ov == v && oi < i)) { v = ov; i = oi; }
            }
            if (lane == 0) s_cur = i;
        }
        __syncthreads();
        cur = s_cur;
    }
}

// ---------------------------------------------------------------------------
// Kernel 2: ball query via V_WMMA_F32_16X16X4_F32. Each wave owns 16
// centroids (A operand, K padded 3->4 with zeros) and sweeps all 8192 points
// in 16-point tiles (B operand). d2 = |s|^2 + |p|^2 - 2*dot (same expansion
// as the reference). Per-row ballots -> manager lanes 0..15 append hits in
// ascending global point index (== top-K smallest indices), pad = S_.
// Inner loop is branchless except the (rare) append loop, so EXEC is all-1s
// at every wmma and no saveexec cascades appear in the hot path.
// ---------------------------------------------------------------------------
__global__ __launch_bounds__(128) void ballq_kernel(const float* __restrict__ xyz,
                                                    const float* __restrict__ sample_xyz,
                                                    int* __restrict__ nbr_idx)
{
    const int lane  = threadIdx.x & 31;
    const int lo    = (lane < 16);                   // low half-wave flag
    const int gwave = blockIdx.x * (blockDim.x >> 5) + (threadIdx.x >> 5);
    const int gpb   = S_ / 16;                       // centroid groups per batch
    const int b      = gwave / gpb;
    const int s_base = (gwave % gpb) * 16;

    // pre-fill this group's 16*K_ index slots with the pad value S_
    int* grp_idx = nbr_idx + ((size_t)b * S_ + s_base) * K_;
    for (int i = lane; i < 16 * K_; i += 32) grp_idx[i] = S_;

    // A operand: centroid (x,y) on lanes 0-15, (z,0) on lanes 16-31 (branchless)
    const int cm = lane & 15;
    const float* cp = sample_xyz + ((size_t)b * S_ + s_base + cm) * 3;
    float a0 = cp[lo ? 0 : 2];
    float am = cp[1];
    float a1 = lo ? am : 0.0f;
    v2f A; A[0] = a0; A[1] = a1;

    // |s|^2 per centroid; pre-broadcast the 8 row norms this lane-half needs
    float part = a0 * a0 + a1 * a1;
    float sn2  = part + __shfl_xor(part, 16);        // lane l: |centroid l%16|^2
    const int hi8 = lo ? 0 : 8;
    float sn2_r[8];
    #pragma unroll
    for (int r = 0; r < 8; ++r) sn2_r[r] = __shfl(sn2, r + hi8);

    int  cnt   = 0;                                  // per-manager-lane hit count
    int* myrow = grp_idx + cm * K_;                  // manager's output row
    const float* pbase = xyz + (size_t)b * N_ * 3;
    const int boff = lo ? 0 : 2;                     // branchless B-load offset
    const unsigned msh = (unsigned)((cm & 8) << 1);  // 0 or 16
    const unsigned mlim = lo ? 0xffffu : 0u;         // lanes >=16 never append
    const int m7 = cm & 7;

    for (int p = 0; p < N_; p += 16) {
        // B operand: point (x,y) on lanes 0-15, (z,0) on lanes 16-31 (branchless)
        const float* pp = pbase + (size_t)(p + cm) * 3;
        float b0 = pp[boff];
        float bm = pp[1];
        float b1 = lo ? bm : 0.0f;
        v2f Bv; Bv[0] = b0; Bv[1] = b1;

        float pq  = b0 * b0 + b1 * b1;
        float pn2 = pq + __shfl_xor(pq, 16);         // lane l: |point p+l%16|^2

        v8f Cz = {};
        // (neg_a, A, neg_b, B, c_mod, C, reuse_a, reuse_b)
        v8f D = __builtin_amdgcn_wmma_f32_16x16x4_f32(
            false, A, false, Bv, (short)0, Cz, false, false);

        unsigned bl0, bl1, bl2, bl3, bl4, bl5, bl6, bl7;
        bl0 = (unsigned)__ballot(sn2_r[0] + pn2 - 2.0f * D[0] < RADIUS2);
        bl1 = (unsigned)__ballot(sn2_r[1] + pn2 - 2.0f * D[1] < RADIUS2);
        bl2 = (unsigned)__ballot(sn2_r[2] + pn2 - 2.0f * D[2] < RADIUS2);
        bl3 = (unsigned)__ballot(sn2_r[3] + pn2 - 2.0f * D[3] < RADIUS2);
        bl4 = (unsigned)__ballot(sn2_r[4] + pn2 - 2.0f * D[4] < RADIUS2);
        bl5 = (unsigned)__ballot(sn2_r[5] + pn2 - 2.0f * D[5] < RADIUS2);
        bl6 = (unsigned)__ballot(sn2_r[6] + pn2 - 2.0f * D[6] < RADIUS2);
        bl7 = (unsigned)__ballot(sn2_r[7] + pn2 - 2.0f * D[7] < RADIUS2);

        // branchless OR-of-selects: each term is cndmask(SGPR, inline 0)
        unsigned w = ((m7 == 0) ? bl0 : 0u) | ((m7 == 1) ? bl1 : 0u)
                   | ((m7 == 2) ? bl2 : 0u) | ((m7 == 3) ? bl3 : 0u)
                   | ((m7 == 4) ? bl4 : 0u) | ((m7 == 5) ? bl5 : 0u)
                   | ((m7 == 6) ? bl6 : 0u) | ((m7 == 7) ? bl7 : 0u);
        unsigned mask = (w >> msh) & mlim;           // row cm's 16 hit bits

        while (mask) {                               // rare; ~0.5 hits/tile avg
            int bit = __ffs(mask) - 1;
            mask &= mask - 1;
            if (cnt < K_) myrow[cnt] = p + bit;
            ++cnt;
        }
    }
}

// ---------------------------------------------------------------------------
// Kernel 3: feature gather — the bandwidth roofline (256 MiB written).
// One wave per (b,s,k) row; 256 floats moved as 2 x float4 per lane
// (global_load_b128 / global_store_b128). feat rows stay L2-resident.
// ---------------------------------------------------------------------------
__global__ __launch_bounds__(256) void gather_kernel(const float* __restrict__ feat,
                                                     const int* __restrict__ nbr_idx,
                                                     float* __restrict__ out_feat)
{
    const int lane = threadIdx.x & 31;
    const int row  = blockIdx.x * 8 + (threadIdx.x >> 5);   // over B*S*K rows
    const int b    = row / (S_ * K_);
    const int idx  = nbr_idx[row];                           // pad slots hold S_
    const float4* srcv = (const float4*)(feat + ((size_t)b * N_ + idx) * C_);
    float4*       dstv = (float4*)(out_feat + (size_t)row * C_);
    dstv[lane]      = srcv[lane];
    dstv[lane + 32] = srcv[lane + 32];
}

extern "C" void kernel_launch(void* const* d_in, const int* in_sizes, int n_in,
                              void* d_out, int out_size, void* d_ws, size_t ws_size,
                              hipStream_t stream)
{
    const float* xyz  = (const float*)d_in[0];   // (B,N,3) f32
    const float* feat = (const float*)d_in[1];   // (B,N,C) f32

    float* sample_xyz = (float*)d_out;                        // (B,S,3)
    float* nbr_feat   = (float*)d_out + (size_t)B_ * S_ * 3;  // (B,S,K,C)
    int*   idx_ws     = (int*)d_ws;                           // B*S*K ints (1 MB)

    fps_kernel  <<<B_,                 256, 0, stream>>>(xyz, sample_xyz);
    ballq_kernel<<<(B_ * S_ / 16) / 4, 128, 0, stream>>>(xyz, sample_xyz, idx_ws);
    gather_kernel<<<(B_ * S_ * K_) / 8, 256, 0, stream>>>(feat, idx_ws, nbr_feat);
}